// EarlyFusionModel_45964740001863
// MI455X (gfx1250) — compile-verified
//
#include <hip/hip_runtime.h>
#include <hip/hip_bf16.h>

// ---------------------------------------------------------------------------
// Problem constants (from reference setup_inputs)
// ---------------------------------------------------------------------------
#define B_   4
#define T_   24
#define C_   10
#define H_   128
#define W_   128
#define TC_  365
#define DC_  11        // clim input dim
#define HID_ 64        // hidden
#define DM_  64        // model dim
#define PAD_VALUE (-1000.0f)

#define ROWS (B_ * T_)          // 96 (b,t) rows
#define OUTC (C_ + DM_)         // 74 output channels
#define HW_  (H_ * W_)          // 16384

typedef float v2f __attribute__((ext_vector_type(2)));
typedef float v4f __attribute__((ext_vector_type(4)));
typedef float v8f __attribute__((ext_vector_type(8)));

// ---------------------------------------------------------------------------
// Kernel 1: date-match gather + 2-layer MLP via V_WMMA_F32_16X16X4_F32
//   clim_feat[96,64] = relu(clim[96,11] @ W1 + b1) @ W2 + b2
// One workgroup, 24 waves; wave w handles tile (mt = w/4, nt = w%4).
// f32 WMMA fragment layouts (ISA 7.12.2):
//   A 16x4 : lanes 0-15 -> M=lane, K={0,1} in v[0],v[1]; lanes 16-31 -> K={2,3}
//   B 4x16 : N = lane%16; lanes 0-15 -> K={0,1}; lanes 16-31 -> K={2,3}
//   C/D    : N = lane%16; VGPR v -> M = v + 8*(lane>=16)
// ---------------------------------------------------------------------------
__global__ __launch_bounds__(768)
void clim_mlp_wmma(const float* __restrict__ clim,
                   const int*   __restrict__ dates_sat,
                   const int*   __restrict__ dates_clim,
                   const float* __restrict__ W1, const float* __restrict__ b1,
                   const float* __restrict__ W2, const float* __restrict__ b2,
                   float* __restrict__ feat /* [96*64] */)
{
    __shared__ float h_lds[ROWS * HID_];   // 24 KB

    const int tid  = threadIdx.x;
    const int wave = tid >> 5;
    const int lane = tid & 31;
    const int mt   = wave >> 2;            // 0..5  (M tile of 16 rows)
    const int nt   = wave & 3;             // 0..3  (N tile of 16 cols)
    const int lrow = lane & 15;
    const int hi   = lane >> 4;            // 0: K pair {0,1}; 1: K pair {2,3}
    const int row  = mt * 16 + lrow;       // 0..95
    const int col  = nt * 16 + lrow;       // 0..63
    const int b    = row / T_;
    const int t    = row % T_;

    // ---- per-row date match (first tc with dates_clim == dates_sat) ----
    const int sd = dates_sat[b * T_ + t];
    int idx = -1;
    for (int tc = 0; tc < TC_; ++tc) {
        if (dates_clim[b * TC_ + tc] == sd) { idx = tc; break; }
    }
    const bool has = (idx >= 0);
    // safe base for the gather (masked to PAD below when !has)
    const float* crow = clim + ((size_t)b * TC_ + (has ? idx : 0)) * DC_;

    // A-element fetch: k >= 11 -> 0 (K padding); no date match -> PAD_VALUE.
    // Direct per-lane loads (L2-resident) instead of a register array, so the
    // lane-varying k index doesn't force a 12-way cndmask cascade.
    auto clim_at = [&](int k) -> float {
        if (k >= DC_) return 0.0f;
        return has ? crow[k] : PAD_VALUE;
    };

    // ---- layer 1: [16x12] @ [12x16] in 3 K-chunks of 4 ----
    v8f acc = {};
    #pragma unroll
    for (int c = 0; c < 3; ++c) {
        const int k0 = c * 4 + hi * 2;     // even, lane-varying (hi)
        v2f a; a.x = clim_at(k0); a.y = clim_at(k0 + 1);
        v2f w;
        w.x = (k0     < DC_) ? W1[(k0    ) * HID_ + col] : 0.0f;
        w.y = (k0 + 1 < DC_) ? W1[(k0 + 1) * HID_ + col] : 0.0f;
        acc = __builtin_amdgcn_wmma_f32_16x16x4_f32(
                  false, a, false, w, (short)0, acc, false, false);
    }

    // ---- bias + relu -> LDS (C/D layout: M = v + 8*hi, N = col) ----
    const float bias1 = b1[col];
    #pragma unroll
    for (int v = 0; v < 8; ++v) {
        float x = acc[v] + bias1;
        x = x > 0.0f ? x : 0.0f;
        h_lds[(mt * 16 + v + hi * 8) * HID_ + col] = x;
    }
    __syncthreads();

    // ---- layer 2: [16x64] @ [64x16] in 16 K-chunks of 4 ----
    v8f acc2 = {};
    #pragma unroll
    for (int c = 0; c < 16; ++c) {
        const int k0 = c * 4 + hi * 2;     // even -> 8B-aligned LDS read
        v2f a = *(const v2f*)&h_lds[row * HID_ + k0];   // one ds_load_b64
        v2f w; w.x = W2[(k0    ) * DM_ + col];
               w.y = W2[(k0 + 1) * DM_ + col];
        acc2 = __builtin_amdgcn_wmma_f32_16x16x4_f32(
                   false, a, false, w, (short)0, acc2, false, false);
    }

    const float bias2 = b2[col];
    #pragma unroll
    for (int v = 0; v < 8; ++v)
        feat[(mt * 16 + v + hi * 8) * DM_ + col] = acc2[v] + bias2;
}

// ---------------------------------------------------------------------------
// Kernel 2: streaming concat/broadcast (the bandwidth-critical 465 MB write).
// One block per output channel-plane (128x128 = 4096 float4).
// Non-temporal B128 loads/stores: output is write-once and >> L2 (192 MB).
// Roofline: (465.6 + 62.9) MB @ 23.3 TB/s  ->  ~23 us floor.
// ---------------------------------------------------------------------------
__global__ __launch_bounds__(256)
void fuse_broadcast(const float* __restrict__ sat,
                    const float* __restrict__ feat,
                    float* __restrict__ out)
{
    const int plane = blockIdx.x;          // 0 .. B*T*74-1
    const int ch    = plane % OUTC;
    const int bt    = plane / OUTC;
    const int HW4   = HW_ / 4;             // 4096 float4 per plane

    v4f* __restrict__ op = (v4f*)(out + (size_t)plane * HW_);

    if (ch < C_) {
        const v4f* __restrict__ ip =
            (const v4f*)(sat + ((size_t)bt * C_ + ch) * HW_);
        #pragma unroll 4
        for (int i = threadIdx.x; i < HW4; i += 256) {
            v4f v = __builtin_nontemporal_load(&ip[i]);
            __builtin_nontemporal_store(v, &op[i]);
        }
    } else {
        const float s = feat[bt * DM_ + (ch - C_)];
        v4f v; v.x = s; v.y = s; v.z = s; v.w = s;
        #pragma unroll 4
        for (int i = threadIdx.x; i < HW4; i += 256)
            __builtin_nontemporal_store(v, &op[i]);
    }
}

// ---------------------------------------------------------------------------
// Launch
// ---------------------------------------------------------------------------
extern "C" void kernel_launch(void* const* d_in, const int* in_sizes, int n_in,
                              void* d_out, int out_size, void* d_ws, size_t ws_size,
                              hipStream_t stream)
{
    const float* sat   = (const float*)d_in[0];
    const int*   dsat  = (const int*)  d_in[1];
    const float* clim  = (const float*)d_in[2];
    const int*   dclim = (const int*)  d_in[3];
    const float* W1    = (const float*)d_in[4];
    const float* b1    = (const float*)d_in[5];
    const float* W2    = (const float*)d_in[6];
    const float* b2    = (const float*)d_in[7];
    float* out  = (float*)d_out;
    float* feat = (float*)d_ws;            // 96*64 f32 = 24.6 KB scratch

    clim_mlp_wmma<<<1, 768, 0, stream>>>(clim, dsat, dclim, W1, b1, W2, b2, feat);
    fuse_broadcast<<<B_ * T_ * OUTC, 256, 0, stream>>>(sat, feat, out);
}